// DCNv2_64046552318482
// MI455X (gfx1250) — compile-verified
//
#include <hip/hip_runtime.h>

typedef __attribute__((ext_vector_type(2))) float v2f;
typedef __attribute__((ext_vector_type(8))) float v8f;

#define HW     64
#define C_IN   256
#define OC_OUT 256
#define KTAPS  9
#define NPIX   (4 * 64 * 64)   // 16384
#define OM_CH  27

// workspace layout (floats):
//   om  : NPIX * 27                                  @ 0
//   Wt2 : 9 * 256 * 256  (tap, k/4, pair, oc, q)     @ OM_F   (WMMA-interleaved)
//   owT : 9 * 27 * 256   (tap, m, c)                 @ OM_F + WT_F
#define OM_F  (NPIX * OM_CH)            // 442368
#define WT_F  (KTAPS * C_IN * OC_OUT)   // 589824
#define OWT_F (KTAPS * OM_CH * C_IN)    // 62208

// ---------------------------------------------------------------------------
// Kernel 0: repack weights.
//   dcn_w (OC,C,3,3)  -> Wt2[tap][k>>2][(k>>1)&1][oc][k&1]
//     so that WMMA lane (pair p, oc n) loads its (k, k+1) B pair as one b64.
//   off_w (3,3,C,27)  -> owT[tap][m][c]  (contiguous channels for dot products)
// ---------------------------------------------------------------------------
__global__ void dcn_transpose_weights(const float* __restrict__ dcn_w,
                                      const float* __restrict__ off_w,
                                      float* __restrict__ Wt2,
                                      float* __restrict__ owT) {
    int idx = blockIdx.x * blockDim.x + threadIdx.x;
    if (idx < WT_F) {
        int oc  = idx % OC_OUT;
        int k   = (idx / OC_OUT) % C_IN;
        int tap = idx / (OC_OUT * C_IN);
        int dst = tap * (C_IN * OC_OUT)
                + (k >> 2) * (4 * OC_OUT)
                + ((k >> 1) & 1) * (2 * OC_OUT)
                + oc * 2
                + (k & 1);
        Wt2[dst] = dcn_w[(oc * C_IN + k) * KTAPS + tap];
    }
    if (idx < OWT_F) {
        int c   = idx % C_IN;
        int m   = (idx / C_IN) % OM_CH;
        int tap = idx / (C_IN * OM_CH);
        owT[idx] = off_w[(tap * C_IN + c) * OM_CH + m];
    }
}

// ---------------------------------------------------------------------------
// Kernel 1: 3x3 SAME conv producing the 27-channel offset/mask map (pre-sigmoid).
// ---------------------------------------------------------------------------
__global__ void dcn_offset_conv(const float* __restrict__ x,     // (N,64,64,256)
                                const float* __restrict__ owT,   // (9,27,256)
                                const float* __restrict__ off_b, // (27)
                                float* __restrict__ om) {        // (NPIX,27)
    int idx = blockIdx.x * blockDim.x + threadIdx.x;
    if (idx >= NPIX * OM_CH) return;
    int m   = idx % OM_CH;
    int pix = idx / OM_CH;
    int ox  = pix % HW;
    int oy  = (pix / HW) % HW;
    int n   = pix / (HW * HW);

    float acc = off_b[m];
    for (int tap = 0; tap < KTAPS; ++tap) {
        int ky = tap / 3 - 1, kx = tap % 3 - 1;
        int yy = oy + ky, xx = ox + kx;
        if (yy < 0 || yy >= HW || xx < 0 || xx >= HW) continue;
        const float4* xp = (const float4*)(x + (((size_t)n * HW + yy) * HW + xx) * C_IN);
        const float4* wp = (const float4*)(owT + ((size_t)tap * OM_CH + m) * C_IN);
        #pragma unroll 4
        for (int c4 = 0; c4 < C_IN / 4; ++c4) {
            float4 xv = xp[c4], wv = wp[c4];
            acc += xv.x * wv.x + xv.y * wv.y + xv.z * wv.z + xv.w * wv.w;
        }
    }
    om[idx] = acc;
}

// ---------------------------------------------------------------------------
// Kernel 2: fused deformable sampling + GEMM via V_WMMA_F32_16X16X4_F32.
// Workgroup = 32 output pixels x 256 OCs; 8 waves; each wave owns a 32x32
// output block (4 accumulator tiles -> 4 independent WMMA chains, each A/B
// operand reused twice). K = 9 taps x 256 channels, stepped 4 at a time.
// ---------------------------------------------------------------------------
__device__ __forceinline__ float4 dcn_load4(const float* __restrict__ x,
                                            int n, int yy, int xx, int c) {
    // pad_x coordinates: valid data occupies rows/cols [1,64]; rest is zero pad
    if (yy < 1 || yy > HW || xx < 1 || xx > HW) {
        float4 z; z.x = z.y = z.z = z.w = 0.0f; return z;
    }
    return *(const float4*)(x + (((size_t)n * HW + (yy - 1)) * HW + (xx - 1)) * C_IN + c);
}

#define LDS_STRIDE 260   // 260 % 64 == 4 -> conflict-free b64 A-operand reads

__global__ __launch_bounds__(256) void dcn_main(const float* __restrict__ x,
                                                const float* __restrict__ om,
                                                const float* __restrict__ Wt2,
                                                float* __restrict__ out) {
    __shared__ float lds[32 * LDS_STRIDE];

    const int tid     = threadIdx.x;
    const int pixbase = blockIdx.x * 32;

    // --- sampling-phase assignment: 8 threads per pixel, 32 channels each
    const int sp    = tid & 31;
    const int cbase = (tid >> 5) * 32;
    const int pix   = pixbase + sp;
    const int ox    = pix % HW;
    const int oy    = (pix / HW) % HW;
    const int n     = pix / (HW * HW);

    // --- WMMA-phase assignment
    const int wave   = tid >> 5;          // 0..7
    const int nb     = wave * 32;         // OC block base (two 16-wide subtiles)
    const int lane_n = tid & 15;          // N (and A-row M) index within tile
    const int phi    = (tid & 16) >> 4;   // K-pair select: lanes 0-15 -> {0,1}
    const int koff   = phi * 2;           //                lanes 16-31 -> {2,3}

    v8f acc00 = {}, acc01 = {}, acc10 = {}, acc11 = {};

    for (int tap = 0; tap < KTAPS; ++tap) {
        // ---- per-pixel offset / mask (redundant across the 8 threads of a pixel)
        const float offy = om[pix * OM_CH + 2 * tap];
        const float offx = om[pix * OM_CH + 2 * tap + 1];
        const float msk  = 1.0f / (1.0f + __expf(-om[pix * OM_CH + 18 + tap]));
        float py = (float)(oy + 1) + (float)(tap / 3) - 1.0f + offy;
        float px = (float)(ox + 1) + (float)(tap % 3) - 1.0f + offx;
        py = fminf(fmaxf(py, 0.0f), 65.0f);
        px = fminf(fmaxf(px, 0.0f), 65.0f);
        const float y1 = floorf(py), x1 = floorf(px);
        const float ly = py - y1,   lx = px - x1;
        const float hy = 1.0f - ly, hx = 1.0f - lx;
        const int y1i = (int)y1, x1i = (int)x1;
        const int y2i = y1i + 1, x2i = x1i + 1;
        const float w11 = hy * hx * msk, w12 = hy * lx * msk;
        const float w21 = ly * hx * msk, w22 = ly * lx * msk;

        __syncthreads();   // previous tap's A-operand reads complete
        #pragma unroll
        for (int cc = 0; cc < 32; cc += 4) {
            const int c = cbase + cc;
            float4 v11 = dcn_load4(x, n, y1i, x1i, c);
            float4 v12 = dcn_load4(x, n, y1i, x2i, c);
            float4 v21 = dcn_load4(x, n, y2i, x1i, c);
            float4 v22 = dcn_load4(x, n, y2i, x2i, c);
            float4 r;
            r.x = w11 * v11.x + w12 * v12.x + w21 * v21.x + w22 * v22.x;
            r.y = w11 * v11.y + w12 * v12.y + w21 * v21.y + w22 * v22.y;
            r.z = w11 * v11.z + w12 * v12.z + w21 * v21.z + w22 * v22.z;
            r.w = w11 * v11.w + w12 * v12.w + w21 * v21.w + w22 * v22.w;
            *(float4*)&lds[sp * LDS_STRIDE + c] = r;
        }
        __syncthreads();   // tile ready for WMMA

        // ---- GEMM over this tap's 256-channel K slice
        const float* __restrict__ wtap = Wt2 + (size_t)tap * C_IN * OC_OUT;
        #pragma unroll 4
        for (int kk = 0; kk < C_IN; kk += 4) {
            // A operands (16x4 f32): lane holds rows lane_n and 16+lane_n
            v2f a0 = *(const v2f*)&lds[lane_n * LDS_STRIDE + kk + koff];
            v2f a1 = *(const v2f*)&lds[(16 + lane_n) * LDS_STRIDE + kk + koff];
            // B operands: interleaved layout -> one b64 per lane per subtile
            const float* bb = wtap + (size_t)(kk >> 2) * (4 * OC_OUT) + phi * (2 * OC_OUT);
            v2f b0 = *(const v2f*)&bb[(nb + lane_n) * 2];
            v2f b1 = *(const v2f*)&bb[(nb + 16 + lane_n) * 2];
            acc00 = __builtin_amdgcn_wmma_f32_16x16x4_f32(
                false, a0, false, b0, (short)0, acc00, false, false);
            acc01 = __builtin_amdgcn_wmma_f32_16x16x4_f32(
                false, a0, false, b1, (short)0, acc01, false, false);
            acc10 = __builtin_amdgcn_wmma_f32_16x16x4_f32(
                false, a1, false, b0, (short)0, acc10, false, false);
            acc11 = __builtin_amdgcn_wmma_f32_16x16x4_f32(
                false, a1, false, b1, (short)0, acc11, false, false);
        }
    }

    // ---- store C/D: VGPR v, lanes 0-15 -> M=v, lanes 16-31 -> M=v+8
    const int mhi = (tid & 16) >> 1;   // 0 or 8
    #pragma unroll
    for (int v = 0; v < 8; ++v) {
        const size_t row0 = (size_t)(pixbase + v + mhi) * OC_OUT;
        const size_t row1 = (size_t)(pixbase + 16 + v + mhi) * OC_OUT;
        out[row0 + nb + lane_n]      = acc00[v];
        out[row0 + nb + 16 + lane_n] = acc01[v];
        out[row1 + nb + lane_n]      = acc10[v];
        out[row1 + nb + 16 + lane_n] = acc11[v];
    }
}

// ---------------------------------------------------------------------------
extern "C" void kernel_launch(void* const* d_in, const int* in_sizes, int n_in,
                              void* d_out, int out_size, void* d_ws, size_t ws_size,
                              hipStream_t stream) {
    const float* x      = (const float*)d_in[0];  // (4,64,64,256)
    const float* off_w  = (const float*)d_in[1];  // (3,3,256,27)
    const float* off_b  = (const float*)d_in[2];  // (27)
    const float* dcn_w  = (const float*)d_in[3];  // (256,256,3,3)
    float* out          = (float*)d_out;          // (4,64,64,256)

    float* ws  = (float*)d_ws;
    float* om  = ws;                  // NPIX*27
    float* Wt2 = ws + OM_F;           // 9*256*256 (interleaved)
    float* owT = ws + OM_F + WT_F;    // 9*27*256

    (void)in_sizes; (void)n_in; (void)out_size; (void)ws_size;

    // Kernel 0: weight repack
    {
        int total = WT_F;  // covers OWT_F too (OWT_F < WT_F)
        dcn_transpose_weights<<<(total + 255) / 256, 256, 0, stream>>>(dcn_w, off_w, Wt2, owT);
    }
    // Kernel 1: offset/mask conv
    {
        int total = NPIX * OM_CH;
        dcn_offset_conv<<<(total + 255) / 256, 256, 0, stream>>>(x, owT, off_b, om);
    }
    // Kernel 2: fused sampling + WMMA GEMM
    dcn_main<<<NPIX / 32, 256, 0, stream>>>(x, om, Wt2, out);
}